// GraphGather_56968446214218
// MI455X (gfx1250) — compile-verified
//
#include <hip/hip_runtime.h>
#include <hip/hip_bf16.h>

#define BATCH      4096
#define D_FEAT     128
#define TILE_ROWS  32      // 32 rows x 128 f32 = 16 KB tile
#define C4_PER_ROW (D_FEAT / 4)   // 32 float4 per row

// ---------------------------------------------------------------------------
// gfx1250 async global -> LDS copy (ASYNCcnt-tracked), inline asm (portable
// across ROCm 7.2 / amdgpu-toolchain). Non-temporal: the stream is read once.
//   global_load_async_to_lds_b128  vdst(LDS byte addr), vaddr(64b global), off
// LDS byte address = low 32 bits of the generic pointer to __shared__.
// ---------------------------------------------------------------------------
__device__ __forceinline__ void async_load_f4(void* lds_dst, const void* gsrc) {
    unsigned lds_off = (unsigned)(unsigned long long)lds_dst;
    asm volatile("global_load_async_to_lds_b128 %0, %1, off th:TH_LOAD_NT"
                 :
                 : "v"(lds_off), "v"(gsrc)
                 : "memory");
}

__device__ __forceinline__ void wait_async0() {
    asm volatile("s_wait_asynccnt 0" ::: "memory");
}

// ---------------------------------------------------------------------------
// Kernel 1: membership is sorted -> segment b starts at lower_bound(mem, b).
// starts[BATCH] = n.  (BATCH+1 threads, 21-step binary search each.)
// ---------------------------------------------------------------------------
__global__ void seg_starts_kernel(const int* __restrict__ mem,
                                  int* __restrict__ starts,
                                  int n, int batch) {
    int b = blockIdx.x * blockDim.x + threadIdx.x;
    if (b > batch) return;
    if (b == batch) { starts[b] = n; return; }
    int lo = 0, hi = n;
    while (lo < hi) {
        int mid = (lo + hi) >> 1;
        if (mem[mid] < b) lo = mid + 1; else hi = mid;
    }
    starts[b] = lo;
}

// ---------------------------------------------------------------------------
// Stage one 32x128 tile: 1024 float4 elements, 4 async b128 copies/thread.
// Guards are wave-uniform (e>>5 constant across a wave32), so tails cost a
// scalar branch, not divergence.
// ---------------------------------------------------------------------------
__device__ __forceinline__ void issue_tile(const float4* __restrict__ src,
                                           int r0, int len, int t,
                                           float* __restrict__ buf) {
    const int rows = min(TILE_ROWS, len - r0);
    #pragma unroll
    for (int k = 0; k < 4; ++k) {
        const int e = t + k * 256;           // tile element = row*32 + c4
        if ((e >> 5) < rows) {
            async_load_f4(&buf[e * 4], src + (size_t)r0 * C4_PER_ROW + e);
        }
    }
}

// ---------------------------------------------------------------------------
// Kernel 2: one block per segment. Double-buffered async staging: tile i+1
// streams global->LDS while tile i is consumed from LDS. One s_wait_asynccnt
// + one barrier per 16 KB tile.
// ---------------------------------------------------------------------------
__global__ __launch_bounds__(256)
void seg_reduce_kernel(const float* __restrict__ feat,
                       const int* __restrict__ starts,
                       float* __restrict__ out) {
    __shared__ float  tile[2][TILE_ROWS * D_FEAT];   // 2 x 16 KB staging
    __shared__ float4 s_sum[256];                    // 4 KB
    __shared__ float4 s_max[256];                    // 4 KB

    const int b  = blockIdx.x;
    const int t  = threadIdx.x;
    const int c4 = t & 31;   // float4 column (features 4*c4 .. 4*c4+3)
    const int rg = t >> 5;   // row group 0..7 (rows 4*rg .. 4*rg+3 of tile)

    const int start = starts[b];
    const int len   = starts[b + 1] - start;
    const int ntile = (len + TILE_ROWS - 1) / TILE_ROWS;

    const float NEG_INF = -__builtin_inff();
    float4 sum4 = make_float4(0.f, 0.f, 0.f, 0.f);
    float4 max4 = make_float4(NEG_INF, NEG_INF, NEG_INF, NEG_INF);

    // Segment rows are contiguous: float4 index = (start + row)*32 + c4.
    const float4* __restrict__ src =
        (const float4*)feat + (size_t)start * C4_PER_ROW;

    if (ntile > 0) issue_tile(src, 0, len, t, tile[0]);

    for (int i = 0; i < ntile; ++i) {
        wait_async0();       // my wave's copies (incl. tile i) landed
        __syncthreads();     // all waves: tile i ready AND tile i-1 consumed

        if (i + 1 < ntile)   // prefetch next tile while consuming this one
            issue_tile(src, (i + 1) * TILE_ROWS, len, t, tile[(i + 1) & 1]);

        const float* buf = tile[i & 1];
        const int rows = min(TILE_ROWS, len - i * TILE_ROWS);
        #pragma unroll
        for (int j = 0; j < 4; ++j) {
            const int row = rg * 4 + j;
            if (row < rows) {
                const float4 v = *(const float4*)&buf[(row * 32 + c4) * 4];
                sum4.x += v.x; sum4.y += v.y; sum4.z += v.z; sum4.w += v.w;
                max4.x = fmaxf(max4.x, v.x); max4.y = fmaxf(max4.y, v.y);
                max4.z = fmaxf(max4.z, v.z); max4.w = fmaxf(max4.w, v.w);
            }
        }
    }

    s_sum[t] = sum4;
    s_max[t] = max4;
    __syncthreads();

    // Fold the 8 row-groups (strides 128,64,32 keep the column c4 fixed).
    #pragma unroll
    for (int gap = 128; gap >= 32; gap >>= 1) {
        if (t < gap) {
            float4 a = s_sum[t], c = s_sum[t + gap];
            a.x += c.x; a.y += c.y; a.z += c.z; a.w += c.w;
            s_sum[t] = a;
            float4 m = s_max[t], n = s_max[t + gap];
            m.x = fmaxf(m.x, n.x); m.y = fmaxf(m.y, n.y);
            m.z = fmaxf(m.z, n.z); m.w = fmaxf(m.w, n.w);
            s_max[t] = m;
        }
        __syncthreads();
    }

    if (t < 32) {
        const float denom = fmaxf((float)len, 1.0f);  // jnp.maximum(counts,1)
        const float4 sm = s_sum[t];
        const float4 mx = s_max[t];
        float* o = out + (size_t)b * (2 * D_FEAT);
        o[4 * t + 0]          = tanhf(sm.x / denom);
        o[4 * t + 1]          = tanhf(sm.y / denom);
        o[4 * t + 2]          = tanhf(sm.z / denom);
        o[4 * t + 3]          = tanhf(sm.w / denom);
        o[D_FEAT + 4 * t + 0] = tanhf(mx.x);          // tanh(-inf) = -1 (empty seg)
        o[D_FEAT + 4 * t + 1] = tanhf(mx.y);
        o[D_FEAT + 4 * t + 2] = tanhf(mx.z);
        o[D_FEAT + 4 * t + 3] = tanhf(mx.w);
    }
}

extern "C" void kernel_launch(void* const* d_in, const int* in_sizes, int n_in,
                              void* d_out, int out_size, void* d_ws, size_t ws_size,
                              hipStream_t stream) {
    const float* feat = (const float*)d_in[0];   // [N_ATOMS, 128] f32
    const int*   mem  = (const int*)d_in[1];     // [N_ATOMS] sorted i32
    float*       out  = (float*)d_out;           // [BATCH, 256] f32
    const int n = in_sizes[1];                   // N_ATOMS

    int* starts = (int*)d_ws;                    // (BATCH+1) ints of scratch

    seg_starts_kernel<<<(BATCH + 1 + 255) / 256, 256, 0, stream>>>(
        mem, starts, n, BATCH);
    seg_reduce_kernel<<<BATCH, 256, 0, stream>>>(feat, starts, out);
}